// AnchorHeadBase_23089744183886
// MI455X (gfx1250) — compile-verified
//
#include <hip/hip_runtime.h>
#include <stdint.h>
#include <math.h>

// Problem constants (from reference)
#define BATCH     4
#define CIN       512
#define NXD       200
#define NYD       176
#define S_SP      (NXD * NYD)        // 35200 spatial slots
#define A_ANCH    6                  // 1*3*2
#define NCLASS    4
#define CODE      7
#define O_CLS     (NCLASS * A_ANCH)  // 24
#define O_REG     (CODE * A_ANCH)    // 42
#define MPOS      512
#define MNEG      4096
#define MCLS      (MPOS + MNEG)      // 4608
#define CLS_TILES (MCLS / 16)        // 288
#define REG_TILES (MPOS / 16)        // 32

typedef float v2f __attribute__((ext_vector_type(2)));
typedef float v8f __attribute__((ext_vector_type(8)));

// ---------------------------------------------------------------------------
// Classification branch: 288 tiles of 16 samples; GEMM [16 x 512] x [512 x 32]
// (24 valid output rows, padded to 2 WMMA N-tiles), then log-softmax loss.
// ---------------------------------------------------------------------------
__global__ __launch_bounds__(32) void cls_loss_kernel(
    const float* __restrict__ inputs, const float* __restrict__ w_cls,
    const float* __restrict__ b_cls, const int64_t* __restrict__ pos_ids,
    const int64_t* __restrict__ neg_ids, const int64_t* __restrict__ cls_labels,
    float* __restrict__ partial)
{
    __shared__ float ldsA[16 * CIN];    // gathered A tile (16 samples x 512 ch)
    __shared__ float ldsD[16 * 32];     // D tile (16 samples x 32 outputs)
    __shared__ int   ldsBase[16];
    __shared__ int   ldsJ[16];
    __shared__ float ldsRed[16];

    const int lane = threadIdx.x;
    const int tile = blockIdx.x;

    // lanes 0..15: decode the 16 sample ids of this tile
    if (lane < 16) {
        int s = tile * 16 + lane;
        int64_t b, n;
        if (s < MNEG) { b = neg_ids[2 * s];            n = neg_ids[2 * s + 1]; }
        else          { int p = s - MNEG;
                        b = pos_ids[2 * p];            n = pos_ids[2 * p + 1]; }
        ldsJ[lane]    = (int)(n / S_SP);                       // anchor sub-row
        int sp        = (int)(n % S_SP);                       // spatial slot
        ldsBase[lane] = (int)b * (CIN * S_SP) + sp;
    }
    __syncthreads();

    // Stage A tile: per sample, channel vector is strided by S_SP in global mem
    for (int m = 0; m < 16; ++m) {
        int base = ldsBase[m];
        for (int c = lane; c < CIN; c += 32)
            ldsA[m * CIN + c] = inputs[(size_t)base + (size_t)c * S_SP];
    }
    __syncthreads();

    const int mrow  = lane & 15;   // A row / D column index held by this lane
    const int khalf = lane >> 4;   // which K-pair this lane holds
    const int ncol  = lane & 15;   // B column (output) index

    const float* w0 = w_cls + (size_t)ncol * CIN;                 // rows 0..15
    const int    r1 = ncol + 16;
    const float* w1 = w_cls + (size_t)(r1 < O_CLS ? r1 : 0) * CIN;
    const float  m1 = (r1 < O_CLS) ? 1.0f : 0.0f;                 // pad rows -> 0

    v8f acc0 = {}; v8f acc1 = {};
    for (int k0 = 0; k0 < CIN; k0 += 4) {
        const int kk = k0 + khalf * 2;
        v2f a = *(const v2f*)&ldsA[mrow * CIN + kk];
        v2f b0; b0.x = w0[kk];      b0.y = w0[kk + 1];
        v2f b1; b1.x = w1[kk] * m1; b1.y = w1[kk + 1] * m1;
        acc0 = __builtin_amdgcn_wmma_f32_16x16x4_f32(false, a, false, b0,
                                                     (short)0, acc0, false, false);
        acc1 = __builtin_amdgcn_wmma_f32_16x16x4_f32(false, a, false, b1,
                                                     (short)0, acc1, false, false);
    }

    // D layout: VGPR v, lanes 0-15 -> M=v, lanes 16-31 -> M=v+8; N = lane%16
#pragma unroll
    for (int v = 0; v < 8; ++v) {
        int mo = v + 8 * khalf;
        ldsD[mo * 32 + ncol]      = acc0[v];
        ldsD[mo * 32 + 16 + ncol] = acc1[v];
    }
    __syncthreads();

    // Per-sample cross-entropy over the 4 selected logits
    if (lane < 16) {
        int s   = tile * 16 + lane;
        int j   = ldsJ[lane];
        int lab = (int)cls_labels[s];
        float lg[NCLASS];
#pragma unroll
        for (int ci = 0; ci < NCLASS; ++ci) {
            int o = ci * A_ANCH + j;
            lg[ci] = ldsD[lane * 32 + o] + b_cls[o];
        }
        float mx = fmaxf(fmaxf(lg[0], lg[1]), fmaxf(lg[2], lg[3]));
        float se = 0.0f;
#pragma unroll
        for (int ci = 0; ci < NCLASS; ++ci) se += expf(lg[ci] - mx);
        ldsRed[lane] = -(lg[lab] - mx - logf(se));
    }
    __syncthreads();
    if (lane == 0) {
        float acc = 0.0f;
        for (int i = 0; i < 16; ++i) acc += ldsRed[i];
        partial[tile] = acc;
    }
}

// ---------------------------------------------------------------------------
// Regression branch: 32 tiles of 16 pos samples; GEMM [16x512] x [512x48]
// (42 valid rows, 3 WMMA N-tiles), then smooth-L1 loss vs reg_labels.
// ---------------------------------------------------------------------------
__global__ __launch_bounds__(32) void reg_loss_kernel(
    const float* __restrict__ inputs, const float* __restrict__ w_reg,
    const float* __restrict__ b_reg, const int64_t* __restrict__ pos_ids,
    const float* __restrict__ reg_labels, float* __restrict__ partial)
{
    __shared__ float ldsA[16 * CIN];
    __shared__ float ldsD[16 * 48];
    __shared__ int   ldsBase[16];
    __shared__ int   ldsJ[16];
    __shared__ float ldsRed[16];

    const int lane = threadIdx.x;
    const int tile = blockIdx.x;

    if (lane < 16) {
        int s = tile * 16 + lane;
        int64_t b = pos_ids[2 * s];
        int64_t n = pos_ids[2 * s + 1];
        ldsJ[lane]    = (int)(n / S_SP);
        int sp        = (int)(n % S_SP);
        ldsBase[lane] = (int)b * (CIN * S_SP) + sp;
    }
    __syncthreads();

    for (int m = 0; m < 16; ++m) {
        int base = ldsBase[m];
        for (int c = lane; c < CIN; c += 32)
            ldsA[m * CIN + c] = inputs[(size_t)base + (size_t)c * S_SP];
    }
    __syncthreads();

    const int mrow  = lane & 15;
    const int khalf = lane >> 4;
    const int ncol  = lane & 15;

    const float* w0 = w_reg + (size_t)ncol * CIN;
    const float* w1 = w_reg + (size_t)(ncol + 16) * CIN;
    const int    r2 = ncol + 32;
    const float* w2 = w_reg + (size_t)(r2 < O_REG ? r2 : 0) * CIN;
    const float  m2 = (r2 < O_REG) ? 1.0f : 0.0f;

    v8f acc0 = {}; v8f acc1 = {}; v8f acc2 = {};
    for (int k0 = 0; k0 < CIN; k0 += 4) {
        const int kk = k0 + khalf * 2;
        v2f a = *(const v2f*)&ldsA[mrow * CIN + kk];
        v2f b0; b0.x = w0[kk];      b0.y = w0[kk + 1];
        v2f b1; b1.x = w1[kk];      b1.y = w1[kk + 1];
        v2f b2; b2.x = w2[kk] * m2; b2.y = w2[kk + 1] * m2;
        acc0 = __builtin_amdgcn_wmma_f32_16x16x4_f32(false, a, false, b0,
                                                     (short)0, acc0, false, false);
        acc1 = __builtin_amdgcn_wmma_f32_16x16x4_f32(false, a, false, b1,
                                                     (short)0, acc1, false, false);
        acc2 = __builtin_amdgcn_wmma_f32_16x16x4_f32(false, a, false, b2,
                                                     (short)0, acc2, false, false);
    }

#pragma unroll
    for (int v = 0; v < 8; ++v) {
        int mo = v + 8 * khalf;
        ldsD[mo * 48 + ncol]      = acc0[v];
        ldsD[mo * 48 + 16 + ncol] = acc1[v];
        ldsD[mo * 48 + 32 + ncol] = acc2[v];
    }
    __syncthreads();

    if (lane < 16) {
        int s = tile * 16 + lane;
        int j = ldsJ[lane];
        float acc = 0.0f;
#pragma unroll
        for (int ci = 0; ci < CODE; ++ci) {
            int o = ci * A_ANCH + j;
            float pred = ldsD[lane * 48 + o] + b_reg[o];
            float d  = pred - reg_labels[(size_t)s * CODE + ci];
            float ad = fabsf(d);
            acc += (ad < 1.0f) ? 0.5f * d * d : ad - 0.5f;
        }
        ldsRed[lane] = acc;
    }
    __syncthreads();
    if (lane == 0) {
        float acc = 0.0f;
        for (int i = 0; i < 16; ++i) acc += ldsRed[i];
        partial[tile] = acc;
    }
}

// ---------------------------------------------------------------------------
// Final deterministic reduction: fixed-order sums, weighted combine.
// ---------------------------------------------------------------------------
__global__ __launch_bounds__(32) void final_reduce_kernel(
    const float* __restrict__ pc, const float* __restrict__ pr,
    float* __restrict__ out)
{
    if (threadIdx.x == 0 && blockIdx.x == 0) {
        float cs = 0.0f;
        for (int i = 0; i < CLS_TILES; ++i) cs += pc[i];
        float rs = 0.0f;
        for (int i = 0; i < REG_TILES; ++i) rs += pr[i];
        float cls_loss = cs / (float)MCLS;
        float reg_loss = rs / (float)(MPOS * CODE);
        out[0] = 1.0f * cls_loss + 2.0f * reg_loss;
    }
}

extern "C" void kernel_launch(void* const* d_in, const int* in_sizes, int n_in,
                              void* d_out, int out_size, void* d_ws, size_t ws_size,
                              hipStream_t stream) {
    (void)in_sizes; (void)n_in; (void)out_size; (void)ws_size;
    const float*   inputs     = (const float*)d_in[0];
    const float*   w_cls      = (const float*)d_in[1];
    const float*   b_cls      = (const float*)d_in[2];
    const float*   w_reg      = (const float*)d_in[3];
    const float*   b_reg      = (const float*)d_in[4];
    const float*   reg_labels = (const float*)d_in[5];
    const int64_t* pos_ids    = (const int64_t*)d_in[6];
    const int64_t* neg_ids    = (const int64_t*)d_in[7];
    const int64_t* cls_labels = (const int64_t*)d_in[8];

    float* pc = (float*)d_ws;            // 288 cls partials
    float* pr = pc + CLS_TILES;          // 32 reg partials

    cls_loss_kernel<<<CLS_TILES, 32, 0, stream>>>(inputs, w_cls, b_cls,
                                                  pos_ids, neg_ids, cls_labels, pc);
    reg_loss_kernel<<<REG_TILES, 32, 0, stream>>>(inputs, w_reg, b_reg,
                                                  pos_ids, reg_labels, pr);
    final_reduce_kernel<<<1, 32, 0, stream>>>(pc, pr, (float*)d_out);
}